// Seq2SeqLoss_90348932038999
// MI455X (gfx1250) — compile-verified
//
#include <hip/hip_runtime.h>
#include <hip/hip_bf16.h>
#include <stdint.h>

#define TPB   1024
#define NWAVE (TPB / 32)

// ---------- wave32 reductions (CDNA5 is wave32-only) ----------
__device__ __forceinline__ float wave_max(float v) {
#pragma unroll
  for (int m = 16; m >= 1; m >>= 1) v = fmaxf(v, __shfl_xor(v, m, 32));
  return v;
}
__device__ __forceinline__ float wave_sum(float v) {
#pragma unroll
  for (int m = 16; m >= 1; m >>= 1) v += __shfl_xor(v, m, 32);
  return v;
}

// One async b128 lane-copy: copies 16B per lane from the global address saddr plus goff into LDS at lds_addr.
// TH_LOAD_NT: logits are streamed once (524 MB >> 192 MB L2) -> don't thrash
// L2, keep the shared weight vector resident there.
__device__ __forceinline__ void async_cp16_nt(uint32_t lds_addr, uint32_t goff,
                                              uint64_t saddr) {
  asm volatile("global_load_async_to_lds_b128 %0, %1, %2 th:TH_LOAD_NT"
               :: "v"(lds_addr), "v"(goff), "s"(saddr)
               : "memory");
}

// One workgroup per row. Row is async-DMA'd into LDS once (HBM read exactly
// once -> bandwidth-bound optimum), then two LDS passes: exact max, then a
// fused sumexp + weighted-dot + weight-sum pass.
__global__ __launch_bounds__(TPB) void row_loss_kernel(
    const float* __restrict__ logits,
    const int* __restrict__ gold,
    const unsigned char* __restrict__ mask,
    const float* __restrict__ weight,
    float* __restrict__ row_loss,
    int C, float uniform, float conf_minus_uniform)
{
  extern __shared__ float srow[];        // C floats (128 KB for C=32000)
  __shared__ float red0[NWAVE];
  __shared__ float red1[NWAVE];
  __shared__ float red2[NWAVE];
  __shared__ float red3[NWAVE];

  const int row  = blockIdx.x;
  const int tid  = threadIdx.x;
  const int lane = tid & 31;
  const int wid  = tid >> 5;

  const float* rowp = logits + (size_t)row * (size_t)C;
  const uint64_t gbase    = (uint64_t)(uintptr_t)rowp;           // uniform SGPR base
  const uint32_t lds_base = (uint32_t)(uintptr_t)(&srow[0]);     // low 32b = LDS byte offset

  // ---- async global -> LDS copy of the whole row, manually unrolled x4 ----
  const int nchunk = C >> 2;                 // number of b128 chunks
  {
    int i = tid;
    uint32_t la = lds_base + ((uint32_t)tid << 4);
    uint32_t go = ((uint32_t)tid << 4);
    const uint32_t step1 = (uint32_t)TPB << 4;   // 16 KB
    for (; i + 3 * TPB < nchunk; i += 4 * TPB) {
      async_cp16_nt(la,             go,             gbase);
      async_cp16_nt(la + step1,     go + step1,     gbase);
      async_cp16_nt(la + 2 * step1, go + 2 * step1, gbase);
      async_cp16_nt(la + 3 * step1, go + 3 * step1, gbase);
      la += 4 * step1;
      go += 4 * step1;
    }
    for (; i < nchunk; i += TPB) {
      async_cp16_nt(la, go, gbase);
      la += step1;
      go += step1;
    }
  }
  // scalar tail if C not a multiple of 4
  for (int c = (nchunk << 2) + tid; c < C; c += TPB) srow[c] = rowp[c];

  asm volatile("s_wait_asynccnt 0" ::: "memory");  // my wave's ASYNC copies done
  __syncthreads();                                 // everyone's copies done

  // ---- pass 1: exact row max ----
  float m = -3.402823466e+38f;
#pragma unroll 4
  for (int c = tid; c < C; c += TPB) m = fmaxf(m, srow[c]);
  m = wave_max(m);
  if (lane == 0) red0[wid] = m;
  __syncthreads();
  if (wid == 0) {
    float v = wave_max(red0[lane]);
    if (lane == 0) red0[0] = v;
  }
  __syncthreads();
  const float M = red0[0];

  // ---- pass 2: fused sumexp, weighted dot, weight sum ----
  float se = 0.0f, dot = 0.0f, ws = 0.0f;
#pragma unroll 4
  for (int c = tid; c < C; c += TPB) {
    float x = srow[c];
    float w = weight[c];          // weight stays hot in L2 (NT stream spares it)
    se += __expf(x - M);
    dot = __fmaf_rn(w, x, dot);
    ws += w;
  }
  se  = wave_sum(se);
  dot = wave_sum(dot);
  ws  = wave_sum(ws);
  if (lane == 0) { red1[wid] = se; red2[wid] = dot; red3[wid] = ws; }
  __syncthreads();

  if (tid == 0) {
    float sse = 0.0f, sdot = 0.0f, sws = 0.0f;
#pragma unroll
    for (int i = 0; i < NWAVE; ++i) { sse += red1[i]; sdot += red2[i]; sws += red3[i]; }
    const float LSE = M + __logf(sse);
    const int   g   = gold[row];
    const float xg  = srow[g];
    const float wg  = weight[g];
    // total weighted nll = sws*LSE - sdot ;  gold nll = LSE - xg
    float loss = uniform * (sws * LSE - sdot)
               + conf_minus_uniform * wg * (LSE - xg);
    loss *= mask[row] ? 1.0f : 0.0f;
    row_loss[row] = loss;
  }
}

// Deterministic final reduction (fixed-order tree; no float atomics so the
// graph-replayed result is bit-identical every call).
__global__ __launch_bounds__(TPB) void final_reduce_kernel(
    const float* __restrict__ row_loss,
    const unsigned char* __restrict__ mask,
    float* __restrict__ out, int N)
{
  __shared__ float rl[NWAVE];
  __shared__ float rc[NWAVE];
  const int tid = threadIdx.x, lane = tid & 31, wid = tid >> 5;
  float ls = 0.0f, cs = 0.0f;
  for (int i = tid; i < N; i += TPB) {
    ls += row_loss[i];
    cs += mask[i] ? 1.0f : 0.0f;
  }
  ls = wave_sum(ls);
  cs = wave_sum(cs);
  if (lane == 0) { rl[wid] = ls; rc[wid] = cs; }
  __syncthreads();
  if (tid == 0) {
    float sl = 0.0f, sc = 0.0f;
#pragma unroll
    for (int i = 0; i < NWAVE; ++i) { sl += rl[i]; sc += rc[i]; }
    out[0] = sl / (sc + 1e-8f);
  }
}

extern "C" void kernel_launch(void* const* d_in, const int* in_sizes, int n_in,
                              void* d_out, int out_size, void* d_ws, size_t ws_size,
                              hipStream_t stream) {
  const float*         logits = (const float*)d_in[0];
  const int*           gold   = (const int*)d_in[1];          // jax x64 off -> int32
  const unsigned char* mask   = (const unsigned char*)d_in[2];// bool, 1 byte/elem
  const float*         weight = (const float*)d_in[3];

  const int N = in_sizes[1];   // B*S rows (gold ids count)
  const int C = in_sizes[3];   // classes (weight length)

  const float smoothing  = 0.1f;
  const float uniform    = smoothing / (float)(C - 1);
  const float conf_m_uni = (1.0f - smoothing) - uniform;

  float* row_loss = (float*)d_ws;                 // N floats of scratch
  const size_t shmem = (size_t)C * sizeof(float); // full row in LDS (CDNA5: 320KB/WGP)

  row_loss_kernel<<<N, TPB, shmem, stream>>>(logits, gold, mask, weight,
                                             row_loss, C, uniform, conf_m_uni);
  final_reduce_kernel<<<1, TPB, 0, stream>>>(row_loss, mask, (float*)d_out, N);
}